// OURS_GCN_24610162606708
// MI455X (gfx1250) — compile-verified
//
#include <hip/hip_runtime.h>
#include <math.h>

// ---------------------------------------------------------------------------
// GCN attention layer, restructured around binary adjacency algebra.
// N=1024, DH=DIN=DOUT=64, DP=32, K=16.
// Matrix ops use V_WMMA_F32_16X16X4_F32 (full-precision f32 matrix pipe).
// ---------------------------------------------------------------------------

typedef float v2f __attribute__((ext_vector_type(2)));
typedef float v8f __attribute__((ext_vector_type(8)));

#define NPTS 1024
#define TOPK 16

// ---------------- init: atomic min/max slots ------------------------------
__global__ void OURS_init_minmax(unsigned* mnmx) {
    if (threadIdx.x == 0) {
        mnmx[0] = 0x7F800000u;  // minH = +inf
        mnmx[1] = 0u;           // maxH = 0
        mnmx[2] = 0x7F800000u;  // minP = +inf
        mnmx[3] = 0u;           // maxP = 0
    }
}

// ---------------- squared row norms (scaled) ------------------------------
__global__ void OURS_row_norms(const float* __restrict__ XH,
                               const float* __restrict__ XP,
                               float* __restrict__ nH, float* __restrict__ nP,
                               float invsig) {
    int i = blockIdx.x * blockDim.x + threadIdx.x;
    if (i >= NPTS) return;
    float s = 0.f;
    for (int d = 0; d < 64; ++d) { float v = XH[i * 64 + d] * invsig; s += v * v; }
    nH[i] = s;
    s = 0.f;
    for (int d = 0; d < 32; ++d) { float v = XP[i * 32 + d]; s += v * v; }
    nP[i] = s;
}

// ---------------- Gram tile via f32 WMMA + fused distance -----------------
// One wave (32 threads) computes a 16x16 tile of W = ||X_i - X_j|| (scaled),
// with a running atomic min/max (float bits comparable as uint: W >= 0).
// A-frag / B-frag layout: value index along K = vgpr + 2*(lane>=16), M/N = lane&15.
__global__ __launch_bounds__(32)
void OURS_gram_dist(const float* __restrict__ X, int D, float scale,
                    const float* __restrict__ nrm, float* __restrict__ W,
                    unsigned* __restrict__ mnmx) {
    const int lane = threadIdx.x;
    const int m    = lane & 15;
    const int koff = (lane >> 4) * 2;
    const int bi = blockIdx.y, bj = blockIdx.x;
    const float* Arow = X + (size_t)(bi * 16 + m) * D;
    const float* Brow = X + (size_t)(bj * 16 + m) * D;
    v8f c = {0.f, 0.f, 0.f, 0.f, 0.f, 0.f, 0.f, 0.f};
    for (int k = 0; k < D; k += 4) {
        v2f a, b;
        a.x = Arow[k + koff] * scale;
        a.y = Arow[k + koff + 1] * scale;
        b.x = Brow[k + koff] * scale;
        b.y = Brow[k + koff + 1] * scale;
        c = __builtin_amdgcn_wmma_f32_16x16x4_f32(false, a, false, b,
                                                  (short)0, c, false, false);
    }
    const int hi = lane >> 4;
    const int j  = bj * 16 + m;
    const float nj = nrm[j];
    float lmin = __uint_as_float(0x7F800000u), lmax = 0.f;
#pragma unroll
    for (int r = 0; r < 8; ++r) {
        int i = bi * 16 + r + 8 * hi;
        float d2 = nrm[i] + nj - 2.f * c[r];
        float w  = sqrtf(fmaxf(d2, 0.f));
        W[(size_t)i * NPTS + j] = w;
        lmin = fminf(lmin, w);
        lmax = fmaxf(lmax, w);
    }
    atomicMin(mnmx + 0, __float_as_uint(lmin));
    atomicMax(mnmx + 1, __float_as_uint(lmax));
}

// ---------------- WW = exp(-normH/2) + 0.2*exp(-normP), in place into WH --
__global__ void OURS_ww(float* __restrict__ WH, const float* __restrict__ WP,
                        const unsigned* __restrict__ mnmx) {
    int idx = blockIdx.x * blockDim.x + threadIdx.x;
    float mnH = __uint_as_float(mnmx[0]), mxH = __uint_as_float(mnmx[1]);
    float mnP = __uint_as_float(mnmx[2]), mxP = __uint_as_float(mnmx[3]);
    float u = (WH[idx] - mnH) / (mxH - mnH);
    float v = (WP[idx] - mnP) / (mxP - mnP);
    WH[idx] = expf(-0.5f * u) + 0.2f * expf(-v);
}

// ---------------- per-row top-16 -> bitmask (ties: lowest index) ----------
__global__ __launch_bounds__(256)
void OURS_topk(const float* __restrict__ WW, unsigned* __restrict__ rowmask) {
    __shared__ float row[NPTS];
    __shared__ float rv[256];
    __shared__ int   ri[256];
    __shared__ unsigned mask[32];
    const int r = blockIdx.x, t = threadIdx.x;
    for (int j = t; j < NPTS; j += 256) row[j] = WW[(size_t)r * NPTS + j];
    if (t < 32) mask[t] = 0u;
    __syncthreads();
    for (int it = 0; it < TOPK; ++it) {
        float bv = -3.0e38f;
        int   bi = 0;
        for (int j = t; j < NPTS; j += 256) {
            float v = row[j];
            if (v > bv) { bv = v; bi = j; }   // strict > : lowest index on tie
        }
        rv[t] = bv; ri[t] = bi;
        __syncthreads();
        for (int s = 128; s > 0; s >>= 1) {
            if (t < s) {
                if (rv[t + s] > rv[t] ||
                    (rv[t + s] == rv[t] && ri[t + s] < ri[t])) {
                    rv[t] = rv[t + s];
                    ri[t] = ri[t + s];
                }
            }
            __syncthreads();
        }
        if (t == 0) {
            int j = ri[0];
            row[j] = -3.0e38f;
            mask[j >> 5] |= 1u << (j & 31);
        }
        __syncthreads();
    }
    if (t < 32) rowmask[r * 32 + t] = mask[t];
}

// ---------------- symmetrize + column degree + diagonal -------------------
// Adj[i][j] = rowmask[i][j] | rowmask[j][i]; deg[j] = sum_i Adj[i][j].
__global__ void OURS_degree(const unsigned* __restrict__ rowmask,
                            int* __restrict__ deg, int* __restrict__ diag) {
    int j = blockIdx.x * blockDim.x + threadIdx.x;
    if (j >= NPTS) return;
    const int wj = j >> 5;
    const unsigned bj = 1u << (j & 31);
    int d = 0;
    for (int w = 0; w < 32; ++w) {
        unsigned rw = rowmask[j * 32 + w];  // row j -> transposed contribution
        for (int b = 0; b < 32; ++b) {
            int i = w * 32 + b;
            unsigned b1 = rowmask[i * 32 + wj] & bj;  // Adj row i, col j
            unsigned b2 = rw & (1u << b);             // Adj row j, col i
            d += (b1 | b2) ? 1 : 0;
        }
    }
    deg[j]  = d;
    diag[j] = (rowmask[j * 32 + wj] & bj) ? 1 : 0;
}

// ---------------- C[1024x64] = A[1024x64] @ B (optionally B^T), f32 WMMA ---
__global__ __launch_bounds__(32)
void OURS_gemm64(const float* __restrict__ A, const float* __restrict__ B,
                 float* __restrict__ C, int transB) {
    const int lane = threadIdx.x;
    const int m    = lane & 15;
    const int koff = (lane >> 4) * 2;
    const int bi = blockIdx.y, bj = blockIdx.x;
    v8f c = {0.f, 0.f, 0.f, 0.f, 0.f, 0.f, 0.f, 0.f};
    for (int k = 0; k < 64; k += 4) {
        v2f a, b;
        a.x = A[(bi * 16 + m) * 64 + k + koff];
        a.y = A[(bi * 16 + m) * 64 + k + koff + 1];
        if (transB) {  // B_frag[kk][n] = B[n][kk]
            b.x = B[(bj * 16 + m) * 64 + k + koff];
            b.y = B[(bj * 16 + m) * 64 + k + koff + 1];
        } else {       // B_frag[kk][n] = B[kk][n]
            b.x = B[(k + koff) * 64 + bj * 16 + m];
            b.y = B[(k + koff + 1) * 64 + bj * 16 + m];
        }
        c = __builtin_amdgcn_wmma_f32_16x16x4_f32(false, a, false, b,
                                                  (short)0, c, false, false);
    }
    const int hi = lane >> 4;
#pragma unroll
    for (int r = 0; r < 8; ++r)
        C[(bi * 16 + r + 8 * hi) * 64 + bj * 16 + m] = c[r];
}

// ---------------- closed-form BatchNorm stats over N^2 virtual rows -------
// h0[j,d] = y+pb (Adj=0 rows), h1[j,d] = 2y+pb (Adj=1 rows), counts from deg.
__global__ __launch_bounds__(64)
void OURS_bn_stats(const float* __restrict__ y, const int* __restrict__ deg,
                   const float* __restrict__ psi_b,
                   const float* __restrict__ gamma,
                   float* __restrict__ stats /* [0:64)=A, [64:128)=mu */) {
    int d = threadIdx.x;
    float pb = psi_b[d];
    float s1 = 0.f, s2 = 0.f;
    for (int j = 0; j < NPTS; ++j) {
        float yv = y[j * 64 + d];
        float h0 = yv + pb;
        float h1 = 2.f * yv + pb;
        float dg = (float)deg[j];
        s1 += (float)NPTS * h0 + dg * yv;  // (N-dg)h0 + dg*h1
        s2 += ((float)NPTS - dg) * h0 * h0 + dg * h1 * h1;
    }
    const float invN2 = 1.f / ((float)NPTS * (float)NPTS);
    float mu  = s1 * invN2;
    float var = s2 * invN2 - mu * mu;
    stats[d]      = rsqrtf(var + 1e-5f) * gamma[d];
    stats[64 + d] = mu;
}

// ---------------- fused sigmoid-attention output --------------------------
// out[j,d] = x[j,d]*((deg_j - diag_j)*sigmoid(BN(h1)) + diag_j) + bias[d]
__global__ void OURS_out(const float* __restrict__ x, const float* __restrict__ y,
                         const int* __restrict__ deg, const int* __restrict__ diag,
                         const float* __restrict__ psi_b,
                         const float* __restrict__ beta,
                         const float* __restrict__ bias,
                         const float* __restrict__ stats,
                         float* __restrict__ out) {
    int idx = blockIdx.x * blockDim.x + threadIdx.x;  // 65536
    int j = idx >> 6, d = idx & 63;
    float A  = stats[d], mu = stats[64 + d];
    float h1 = 2.f * y[idx] + psi_b[d];
    float z  = (h1 - mu) * A + beta[d];
    float a1 = 1.f / (1.f + expf(-z));
    float dg = (float)deg[j], di = (float)diag[j];
    float S  = (dg - di) * a1 + di;
    float o  = x[idx] * S + bias[d];
    out[idx]          = o;   // tuple slot 0: out[:, :64] == out
    out[98304 + idx]  = o;   // tuple slot 2: out
}

__global__ void OURS_copy_xp(const float* __restrict__ XP, float* __restrict__ out) {
    int k = blockIdx.x * blockDim.x + threadIdx.x;  // 32768
    out[65536 + k] = XP[k];                         // tuple slot 1: XP
}

// ---------------------------------------------------------------------------
extern "C" void kernel_launch(void* const* d_in, const int* in_sizes, int n_in,
                              void* d_out, int out_size, void* d_ws, size_t ws_size,
                              hipStream_t stream) {
    (void)in_sizes; (void)n_in; (void)out_size; (void)ws_size;
    const float* XH     = (const float*)d_in[0];
    const float* XP     = (const float*)d_in[1];
    const float* XX     = (const float*)d_in[2];
    /* d_in[3] = K (==16, baked into TOPK) */
    const float* weight = (const float*)d_in[4];
    const float* bias   = (const float*)d_in[5];
    const float* psi_w  = (const float*)d_in[6];
    const float* psi_b  = (const float*)d_in[7];
    const float* gamma  = (const float*)d_in[8];
    const float* beta   = (const float*)d_in[9];
    float* out = (float*)d_out;

    // workspace layout (float units)
    float* ws  = (float*)d_ws;
    float* WH  = ws;                       // 1048576  (becomes WW)
    float* WP  = ws + 1048576;             // 1048576
    float* x   = ws + 2097152;             // 65536
    float* y   = ws + 2162688;             // 65536
    float* nH  = ws + 2228224;             // 1024
    float* nP  = ws + 2229248;             // 1024
    float* stats = ws + 2230272;           // 128
    unsigned* mnmx   = (unsigned*)(ws + 2230400);  // 4
    int* deg         = (int*)(ws + 2230404);       // 1024
    int* diag        = (int*)(ws + 2230405 + 1023);// 1024 (= ws+2231428)
    unsigned* rowmask = (unsigned*)(ws + 2232452); // 32768

    const float invsig = (float)(1.0 / (0.1 + 2.220446049250313e-16));

    OURS_init_minmax<<<1, 32, 0, stream>>>(mnmx);
    OURS_row_norms<<<NPTS / 256, 256, 0, stream>>>(XH, XP, nH, nP, invsig);

    dim3 gtiles(64, 64);
    OURS_gram_dist<<<gtiles, 32, 0, stream>>>(XH, 64, invsig, nH, WH, mnmx);
    OURS_gram_dist<<<gtiles, 32, 0, stream>>>(XP, 32, 1.0f,  nP, WP, mnmx + 2);

    OURS_ww<<<(NPTS * NPTS) / 256, 256, 0, stream>>>(WH, WP, mnmx);
    OURS_topk<<<NPTS, 256, 0, stream>>>(WH, rowmask);
    OURS_degree<<<NPTS / 256, 256, 0, stream>>>(rowmask, deg, diag);

    dim3 mtiles(4, 64);
    OURS_gemm64<<<mtiles, 32, 0, stream>>>(XX, weight, x, /*transB=*/0);
    OURS_gemm64<<<mtiles, 32, 0, stream>>>(x, psi_w,  y, /*transB=*/1);

    OURS_bn_stats<<<1, 64, 0, stream>>>(y, deg, psi_b, gamma, stats);
    OURS_out<<<65536 / 256, 256, 0, stream>>>(x, y, deg, diag, psi_b, beta, bias,
                                              stats, out);
    OURS_copy_xp<<<32768 / 256, 256, 0, stream>>>(XP, out);
}